// MA3_PM_53833120088699
// MI455X (gfx1250) — compile-verified
//
#include <hip/hip_runtime.h>
#include <math.h>

// ---------------------------------------------------------------------------
// Hydrological bucket model:
//   Phase 1 (parallel): per-step forcing transforms + 3-head MLP (PET) using
//                       V_WMMA_F32_16X16X4_F32 (exact shape: 16t x 4f x 16h)
//   Phase 2 (serial):   500k-step nonlinear recurrence, single wave,
//                       GLOBAL_LOAD_ASYNC_TO_LDS staging (ASYNCcnt) when avail.
// Workspace: T_STEPS * 8 floats = 16 MB in d_ws.
// ---------------------------------------------------------------------------

#define T_STEPS 500000

typedef __attribute__((ext_vector_type(2))) float v2f;
typedef __attribute__((ext_vector_type(8))) float v8f;
typedef __attribute__((ext_vector_type(4))) int   v4i_t;

#if defined(__has_builtin)
#  if __has_builtin(__builtin_amdgcn_global_load_async_to_lds_b128)
#    define HAS_ASYNC_LOAD_LDS 1
#  else
#    define HAS_ASYNC_LOAD_LDS 0
#  endif
#  if __has_builtin(__builtin_amdgcn_global_store_async_from_lds_b32)
#    define HAS_ASYNC_STORE_LDS 1
#  else
#    define HAS_ASYNC_STORE_LDS 0
#  endif
#  if __has_builtin(__builtin_amdgcn_ds_swizzle)
#    define HAS_DS_SWIZZLE 1
#  else
#    define HAS_DS_SWIZZLE 0
#  endif
#else
#  define HAS_ASYNC_LOAD_LDS 0
#  define HAS_ASYNC_STORE_LDS 0
#  define HAS_DS_SWIZZLE 0
#endif

#if HAS_ASYNC_LOAD_LDS || HAS_ASYNC_STORE_LDS
// Address space is a type qualifier: these give AS-qualified pointee types.
typedef __attribute__((address_space(1))) v4i_t as1_v4i;  // global 128-bit
typedef __attribute__((address_space(3))) v4i_t as3_v4i;  // LDS    128-bit
typedef __attribute__((address_space(1))) int   as1_int;  // global 32-bit
typedef __attribute__((address_space(3))) int   as3_int;  // LDS    32-bit
#endif

// Branchless softplus: max(x,0) + log(1+exp(-|x|)); exp arg <= 0 -> no overflow
__device__ __forceinline__ float softplus_f(float x) {
  return fmaxf(x, 0.0f) + __logf(1.0f + __expf(-fabsf(x)));
}
__device__ __forceinline__ float sigmoid_f(float x) {
  return 1.0f / (1.0f + __expf(-x));
}
__device__ __forceinline__ float relu_f(float x) { return fmaxf(x, 0.0f); }

// XOR butterfly via ds_swizzle_b32 (group-of-32: imm = (xor<<10)|and 0x1F)
#if HAS_DS_SWIZZLE
#  define XOR_ADD(p, immv)                                                   \
     p += __int_as_float(__builtin_amdgcn_ds_swizzle(__float_as_int(p), (immv)))
#else
#  define XOR_ADD(p, immv) p += __shfl_xor((p), ((immv) >> 10) & 31, 32)
#endif

// ---------------------------------------------------------------------------
// Phase 1: one wave handles 16 consecutive timesteps.
//   A (16x4 f32):  lanes 0-15 hold feats[t][0:2], lanes 16-31 feats[t][2:4]
//   B (4x16 f32):  lane n holds W1[k][0..1][n]; lanes 16-31 rows 2,3
//   D (16x16 f32): VGPR r -> timestep r (lanes 0-15) / 8+r (lanes 16-31),
//                  lane&15 -> hidden unit
// feats = [Rn, Deficit, LAI, T_air]
// ---------------------------------------------------------------------------
__global__ void hydro_precompute_kernel(
    const float* __restrict__ forcing,   // (T,12)
    const float* __restrict__ scaled,    // (T,12)
    const float* __restrict__ P_scale,   // (1)
    const float* __restrict__ T_scale,   // (1)
    const float* __restrict__ th,        // (52)
    const float* __restrict__ W1,        // (4,4,16)
    const float* __restrict__ b1,        // (4,16)
    const float* __restrict__ w2,        // (4,16)
    const float* __restrict__ b2,        // (4)
    float* __restrict__ step_data)       // (T,8): rain,snow,fv,mC,mS,pC,pS,pB
{
  const int lane = threadIdx.x & 31;
  const int wave = threadIdx.x >> 5;
  const int tile = blockIdx.x * (blockDim.x >> 5) + wave;
  if (tile >= (T_STEPS / 16)) return;   // wave-uniform: EXEC all-1 inside

  const int t0   = tile * 16;
  const int half = lane >> 4;           // 0: features 0,1   1: features 2,3
  const int l    = lane & 15;
  const int trow = t0 + l;              // timestep owned by this lane (per half)

  const float Ts = T_scale[0];
  const float Ps = P_scale[0];

  const float* fr = forcing + (size_t)trow * 12;
  const float Rn      = fr[0];
  const float Tair    = fr[1] * Ts;
  const float LAI     = fr[2];
  const float Precip  = fr[3] * Ps;
  const float Deficit = fr[4];
  const float Tskin   = scaled[(size_t)trow * 12 + 9] * Ts;

  // A-matrix fragment
  v2f a;
  a.x = half ? LAI  : Rn;
  a.y = half ? Tair : Deficit;

  // Per-step scalar channel (lanes 0-15 store; both halves compute identically)
  {
    const float b_f_veg1  = __expf(th[4]) + 1.0f;
    const float rain_frac = sigmoid_f(softplus_f(th[50]) * (Tair - th[51]));
    const float Rain   = Precip * rain_frac;
    const float Snow   = Precip - Rain;
    const float f_veg  = sigmoid_f(th[5] + b_f_veg1 * LAI);
    const float melt_C = softplus_f(th[44] + th[45] * Tskin) * sigmoid_f(th[46] * Tair);
    const float melt_S = softplus_f(th[47] + th[48] * Tskin) * sigmoid_f(th[49] * Tair);
    float* row = step_data + (size_t)trow * 8;
    if (half == 0) {
      row[0] = Rain; row[1] = Snow; row[2] = f_veg;
      row[3] = melt_C; row[4] = melt_S;
    }
  }

  // Three PET heads (k = 1,2,3), one f32 WMMA each for layer 1
#pragma unroll
  for (int k = 1; k < 4; ++k) {
    const float* Wk = W1 + k * 64;      // [feature][hidden]
    v2f bm;
    bm.x = Wk[(2 * half + 0) * 16 + l];
    bm.y = Wk[(2 * half + 1) * 16 + l];

    const float b1v = b1[k * 16 + l];
    v8f c;
#pragma unroll
    for (int r = 0; r < 8; ++r) c[r] = b1v;  // bias broadcast along timestep dim

    // D(16x16) = A(16x4) x B(4x16) + C  -> v_wmma_f32_16x16x4_f32
    c = __builtin_amdgcn_wmma_f32_16x16x4_f32(
        /*neg_a=*/false, a, /*neg_b=*/false, bm,
        /*c_mod=*/(short)0, c, /*reuse_a=*/false, /*reuse_b=*/false);

    const float w2v = w2[k * 16 + l];
    const float b2v = b2[k];

    float outp[8];
#pragma unroll
    for (int r = 0; r < 8; ++r) {
      // layer 2: dot over hidden units = XOR reduction across the 16-lane half
      float p = softplus_f(c[r]) * w2v;
      XOR_ADD(p, 0x041F);   // xor 1
      XOR_ADD(p, 0x081F);   // xor 2
      XOR_ADD(p, 0x101F);   // xor 4
      XOR_ADD(p, 0x201F);   // xor 8
      outp[r] = softplus_f(p + b2v);
    }
    if (l == 0) {
#pragma unroll
      for (int r = 0; r < 8; ++r)
        step_data[(size_t)(t0 + 8 * half + r) * 8 + 4 + k] = outp[r];
    }
  }
}

// ---------------------------------------------------------------------------
// Phase 2: single wave. Each lane async-DMAs its 32-byte step record straight
// into LDS (GLOBAL_LOAD_ASYNC_TO_LDS_B128, ASYNCcnt); lane 0 runs the serial
// recurrence out of LDS; Q streams back LDS->global asynchronously.
// ---------------------------------------------------------------------------
__global__ void hydro_scan_kernel(const float* __restrict__ step_data, // (T,8)
                                  const float* __restrict__ th,        // (52)
                                  const float* __restrict__ Kp,        // (1)
                                  float* __restrict__ Q)               // (T)
{
  __shared__ float buf[32 * 8];
  __shared__ float qbuf[32];
  const int lane = threadIdx.x;

  // theta-derived constants (uniform)
  const float C_max = __expf(th[12]);
  const float S_max = __expf(th[16]) * Kp[0];
  const float invCmax = 1.0f / C_max;
  const float invSmax = 1.0f / S_max;

  const float eC = __expf(th[21]), sC = __expf(th[22]),
              oC = __expf(th[23]), rC = __expf(th[24]);
  const float wC = eC + sC + oC + rC;
  const float om_C_ET = eC / wC, om_C_S = sC / wC, om_C_O = oC / wC;

  const float eS = __expf(th[25]), sS = __expf(th[26]),
              oS = __expf(th[27]), rS = __expf(th[28]);
  const float wS = eS + sS + oS + rS;
  const float om_S_ET = eS / wS, om_S_S = sS / wS, om_S_O = oS / wS;

  const float eB = __expf(th[29]), oB = __expf(th[30]), rB = __expf(th[31]);
  const float wB = eB + oB + rB;
  const float om_B_ET = eB / wB, om_B_O = oB / wB;

  const float e32 = __expf(th[32]), e34 = __expf(th[34]);
  const float e36 = __expf(th[36]), e38 = __expf(th[38]);
  const float e40 = __expf(th[40]), e42 = __expf(th[42]);
  const float t33 = th[33], t35 = th[35], t37 = th[37], t39 = th[39],
              t41 = th[41], t43 = th[43];

  float C = 0.f, Cs = 0.f, S = 0.f, Ss = 0.f, B = 0.f, D = 0.f;

  for (int base = 0; base < T_STEPS; base += 32) {
    const float* g = step_data + (size_t)(base + lane) * 8;

#if HAS_ASYNC_LOAD_LDS
    // Per-lane async DMA: 32 bytes memory -> LDS, no VGPR data movement.
    __builtin_amdgcn_global_load_async_to_lds_b128(
        (as1_v4i*)g, (as3_v4i*)(buf + lane * 8), /*offset=*/0, /*cpol=*/0);
    __builtin_amdgcn_global_load_async_to_lds_b128(
        (as1_v4i*)g, (as3_v4i*)(buf + lane * 8), /*offset=*/16, /*cpol=*/0);
    asm volatile("s_wait_asynccnt 0x0" ::: "memory");
#else
    const float4* src = (const float4*)g;
    float4 d0 = src[0];
    float4 d1 = src[1];
    ((float4*)buf)[lane * 2 + 0] = d0;
    ((float4*)buf)[lane * 2 + 1] = d1;
#endif
    if (base + 32 < T_STEPS)  // warm caches for next chunk -> global_prefetch_b8
      __builtin_prefetch(step_data + (size_t)(base + 32 + lane) * 8, 0, 0);
    __syncthreads();

    if (lane == 0) {
      for (int i = 0; i < 32; ++i) {
        const float* x = buf + i * 8;
        const float rain = x[0], snow = x[1], fv = x[2];
        const float mCp = x[3], mSp = x[4];
        const float pC = x[5], pS = x[6], pB = x[7];

        Cs += fv * snow;
        const float mC = fminf(Cs, mCp);
        Cs -= mC;
        C += fv * rain + mC;
        const float OF_C = relu_f(C - C_max);
        C -= OF_C;
        const float rel_o = sigmoid_f(t33 + e32 * (C * invCmax));
        const float rel_s = sigmoid_f(t35 + e34 * (C * invCmax));
        const float O_C  = om_C_O * rel_o * C;
        const float Sp_C = om_C_S * rel_s * C;
        const float E_C  = om_C_ET * fminf(pC, C);
        C = relu_f(C - O_C - Sp_C - E_C);

        Ss += (1.0f - fv) * snow;
        const float mS = fminf(Ss, mSp);
        Ss -= mS;
        S += (1.0f - fv) * rain + mS + O_C + OF_C + Sp_C;
        const float OF_S = relu_f(S - S_max);
        S -= OF_S;
        const float rS_o = sigmoid_f(t37 + e36 * (S * invSmax));
        const float rS_s = sigmoid_f(t39 + e38 * (S * invSmax));
        const float O_S  = om_S_O * rS_o * S;
        const float Sp_S = om_S_S * rS_s * S;
        const float E_S  = om_S_ET * fminf(pS, S);
        S = relu_f(S - O_S - Sp_S - E_S);

        B += Sp_S;
        const float relB = sigmoid_f(t41 + e40 * B);
        const float O_B = om_B_O * relB * B;
        const float E_B = om_B_ET * fminf(pB, B);
        B = relu_f(B - O_B - E_B);

        D += OF_S;
        const float relD = sigmoid_f(t43 + e42 * D);
        const float O_D = relD * D;
        D -= O_D;

        qbuf[i] = O_S + O_B + O_D;
      }
    }
    __syncthreads();

#if HAS_ASYNC_STORE_LDS
    // Stream Q back: LDS -> global, async (covered by next chunk's asynccnt wait)
    __builtin_amdgcn_global_store_async_from_lds_b32(
        (as1_int*)(Q + base + lane), (as3_int*)(qbuf + lane),
        /*offset=*/0, /*cpol=*/0);
#else
    Q[base + lane] = qbuf[lane];
#endif
    __syncthreads();   // qbuf/buf reuse fence for next chunk
  }
#if HAS_ASYNC_STORE_LDS
  asm volatile("s_wait_asynccnt 0x0" ::: "memory");
#endif
}

// ---------------------------------------------------------------------------
extern "C" void kernel_launch(void* const* d_in, const int* in_sizes, int n_in,
                              void* d_out, int out_size, void* d_ws, size_t ws_size,
                              hipStream_t stream) {
  const float* forcing = (const float*)d_in[0];
  const float* scaled  = (const float*)d_in[1];
  const float* Pscale  = (const float*)d_in[2];
  const float* Tscale  = (const float*)d_in[3];
  const float* Kp      = (const float*)d_in[4];
  const float* theta   = (const float*)d_in[5];
  const float* W1      = (const float*)d_in[6];
  const float* b1      = (const float*)d_in[7];
  const float* w2      = (const float*)d_in[8];
  const float* b2      = (const float*)d_in[9];

  float* step_data = (float*)d_ws;      // T_STEPS * 8 floats = 16 MB

  const int tiles = T_STEPS / 16;       // 31250 waves, one WMMA-tile each
  const int wavesPerBlock = 8;          // 256 threads
  const int blocks = (tiles + wavesPerBlock - 1) / wavesPerBlock;

  hydro_precompute_kernel<<<blocks, wavesPerBlock * 32, 0, stream>>>(
      forcing, scaled, Pscale, Tscale, theta, W1, b1, w2, b2, step_data);

  hydro_scan_kernel<<<1, 32, 0, stream>>>(step_data, theta, Kp, (float*)d_out);
}